// Net_28406913696565
// MI455X (gfx1250) — compile-verified
//
#include <hip/hip_runtime.h>
#include <math.h>

typedef __attribute__((ext_vector_type(2))) float v2f;
typedef __attribute__((ext_vector_type(8))) float v8f;

#define KK 5
#define NKER 25

__device__ __forceinline__ int imin(int a, int b) { return a < b ? a : b; }

// ---------------- spline basis ----------------
__global__ void basis_kernel(const float* __restrict__ pseudo,
                             float* __restrict__ basis, int* __restrict__ widx, int E) {
  int e = blockIdx.x * blockDim.x + threadIdx.x;
  if (e >= E) return;
  float v0 = pseudo[2 * e + 0] * (KK - 1);
  float v1 = pseudo[2 * e + 1] * (KK - 1);
  int lo0 = (int)floorf(v0); lo0 = lo0 < 0 ? 0 : (lo0 > KK - 2 ? KK - 2 : lo0);
  int lo1 = (int)floorf(v1); lo1 = lo1 < 0 ? 0 : (lo1 > KK - 2 ? KK - 2 : lo1);
  float f0 = v0 - (float)lo0, f1 = v1 - (float)lo1;
  float4 b;
  b.x = (1.f - f0) * (1.f - f1);
  b.y = (1.f - f0) * f1;
  b.z = f0 * (1.f - f1);
  b.w = f0 * f1;
  ((float4*)basis)[e] = b;
  int4 w;
  w.x = lo0 * KK + lo1;
  w.y = lo0 * KK + (lo1 + 1);
  w.z = (lo0 + 1) * KK + lo1;
  w.w = (lo0 + 1) * KK + (lo1 + 1);
  ((int4*)widx)[e] = w;
}

// ---------------- degree count ----------------
__global__ void deg_kernel(const int* __restrict__ dst, float* __restrict__ deg, int E) {
  int e = blockIdx.x * blockDim.x + threadIdx.x;
  if (e < E) atomicAdd(&deg[dst[e]], 1.0f);
}

// ---------------- Z = x @ W[k] via fp32 WMMA (16x16x4) ----------------
// One wave per (16-node tile, kernel k). Branch-free zero padding via clamped
// addresses + selects; v2f (b64) A loads when CIN is a multiple of 4.
template <int CIN, int COUT>
__global__ __launch_bounds__(32) void z_gemm_wmma(const float* __restrict__ x,
                                                  const float* __restrict__ W,
                                                  float* __restrict__ Z) {
  const int blk = blockIdx.x;
  const int k = blk % NKER;
  const int n0 = (blk / NKER) * 16;
  const int lane = threadIdx.x;
  const int lolane = lane & 15;
  const int hi = lane >> 4;  // 0: K pair {0,1}, 1: K pair {2,3}
  constexpr int KSTEPS = (CIN + 3) / 4;

  // A fragments (ISA 7.12.2): lanes 0-15 = rows, VGPR0=K0, VGPR1=K1;
  // lanes 16-31 carry K2,K3 for the same rows.
  v2f a[KSTEPS];
  const float* xr = x + (size_t)(n0 + lolane) * CIN;
#pragma unroll
  for (int s = 0; s < KSTEPS; ++s) {
    const int c0 = s * 4 + 2 * hi;
    if constexpr (CIN % 4 == 0) {
      a[s] = *(const v2f*)(xr + c0);  // contiguous, 8B-aligned
    } else {
      float a0 = xr[imin(c0, CIN - 1)];       // always-valid address
      float a1 = xr[imin(c0 + 1, CIN - 1)];
      a0 = (c0 < CIN) ? a0 : 0.f;             // v_cndmask, no branch
      a1 = (c0 + 1 < CIN) ? a1 : 0.f;
      a[s] = (v2f){a0, a1};
    }
  }

  const float* Wk = W + (size_t)k * CIN * COUT;
#pragma unroll
  for (int ot = 0; ot < COUT / 16; ++ot) {
    const int col = ot * 16 + lolane;
    v8f c = {};
#pragma unroll
    for (int s = 0; s < KSTEPS; ++s) {
      const int r0 = s * 4 + 2 * hi;
      float b0 = Wk[(size_t)imin(r0, CIN - 1) * COUT + col];
      float b1 = Wk[(size_t)imin(r0 + 1, CIN - 1) * COUT + col];
      if constexpr (CIN % 4 != 0) {
        b0 = (r0 < CIN) ? b0 : 0.f;
        b1 = (r0 + 1 < CIN) ? b1 : 0.f;
      }
      v2f bf = (v2f){b0, b1};
      c = __builtin_amdgcn_wmma_f32_16x16x4_f32(false, a[s], false, bf,
                                                (short)0, c, false, false);
    }
    // D layout: lanes 0-15 hold M=0..7 in c[0..7] (N=lane), lanes 16-31 hold M=8..15
    float* Zo = Z + ((size_t)n0 * NKER + k) * COUT + col;
#pragma unroll
    for (int j = 0; j < 8; ++j) {
      int m = j + 8 * hi;
      Zo[(size_t)m * NKER * COUT] = c[j];
    }
  }
}

// ---------------- edge gather + weighted sum + scatter-add ----------------
// One thread per (edge, 2 float4 chunks): 8 b128 gathers + 8 f32 atomics.
template <int COUT>
__global__ void edge_msg(const float* __restrict__ Z, const float* __restrict__ basis,
                         const int* __restrict__ widx, const int* __restrict__ src,
                         const int* __restrict__ dst, float* __restrict__ agg, int E) {
  constexpr int CH = COUT / 4;   // float4 chunks per row
  constexpr int TPE = CH / 2;    // threads per edge (2 chunks each)
  int gid = blockIdx.x * blockDim.x + threadIdx.x;
  if (gid >= E * TPE) return;
  int e = gid / TPE;
  int ch = (gid % TPE) * 2;
  int sn = src[e], dn = dst[e];
  float4 b = ((const float4*)basis)[e];
  int4 w = ((const int4*)widx)[e];
  size_t base = (size_t)sn * NKER;
  const float4* Zv = (const float4*)Z;
  const float4* r0 = Zv + (base + (size_t)w.x) * CH + ch;
  const float4* r1 = Zv + (base + (size_t)w.y) * CH + ch;
  const float4* r2 = Zv + (base + (size_t)w.z) * CH + ch;
  const float4* r3 = Zv + (base + (size_t)w.w) * CH + ch;
  float* ap = agg + (size_t)dn * COUT + ch * 4;
#pragma unroll
  for (int q = 0; q < 2; ++q) {
    float4 z0 = r0[q], z1 = r1[q], z2 = r2[q], z3 = r3[q];
    atomicAdd(ap + 4 * q + 0, b.x * z0.x + b.y * z1.x + b.z * z2.x + b.w * z3.x);
    atomicAdd(ap + 4 * q + 1, b.x * z0.y + b.y * z1.y + b.z * z2.y + b.w * z3.y);
    atomicAdd(ap + 4 * q + 2, b.x * z0.z + b.y * z1.z + b.z * z2.z + b.w * z3.z);
    atomicAdd(ap + 4 * q + 3, b.x * z0.w + b.y * z1.w + b.z * z2.w + b.w * z3.w);
  }
}

// ---------------- normalize + root + bias + ELU ----------------
template <int CIN, int COUT>
__global__ void finalize_elu(const float* __restrict__ agg, const float* __restrict__ deg,
                             const float* __restrict__ xin, const float* __restrict__ root,
                             const float* __restrict__ bias, float* __restrict__ out, int N) {
  int gid = blockIdx.x * blockDim.x + threadIdx.x;
  if (gid >= N * COUT) return;
  int n = gid / COUT, o = gid % COUT;
  float v = agg[gid] / fmaxf(deg[n], 1.0f);
  const float* xr = xin + (size_t)n * CIN;
#pragma unroll
  for (int i = 0; i < CIN; ++i) v += xr[i] * root[i * COUT + o];
  v += bias[o];
  out[gid] = v > 0.f ? v : (expf(v) - 1.f);
}

// ---------------- mean pool per contiguous graph slice ----------------
__global__ void pool_kernel(const float* __restrict__ h, const int* __restrict__ slices,
                            float* __restrict__ g) {
  int grp = blockIdx.x;
  int o = threadIdx.x & 63;
  int sub = threadIdx.x >> 6;  // 0..3
  int n0 = slices[grp], n1 = slices[grp + 1];
  float acc = 0.f;
  for (int n = n0 + sub; n < n1; n += 4) acc += h[(size_t)n * 64 + o];
  __shared__ float red[256];
  red[threadIdx.x] = acc;
  __syncthreads();
  if (sub == 0) {
    float s = red[o] + red[64 + o] + red[128 + o] + red[192 + o];
    float cnt = (float)(n1 - n0);
    g[grp * 64 + o] = s / fmaxf(cnt, 1.0f);
  }
}

// ---------------- FC + log_softmax (one wave32 per graph) ----------------
__global__ __launch_bounds__(32) void fc_logsoftmax(const float* __restrict__ g,
                                                    const float* __restrict__ fcw,
                                                    const float* __restrict__ fcb,
                                                    float* __restrict__ out) {
  int row = blockIdx.x;
  int lane = threadIdx.x;
  float logit = -INFINITY;
  if (lane < 30) {
    float acc = fcb[lane];
    for (int o = 0; o < 64; ++o) acc += g[row * 64 + o] * fcw[o * 30 + lane];
    logit = acc;
  }
  float m = logit;
  for (int off = 16; off; off >>= 1) m = fmaxf(m, __shfl_xor(m, off, 32));
  float ex = (lane < 30) ? expf(logit - m) : 0.f;
  float s = ex;
  for (int off = 16; off; off >>= 1) s += __shfl_xor(s, off, 32);
  if (lane < 30) out[row * 30 + lane] = logit - m - logf(s);
}

extern "C" void kernel_launch(void* const* d_in, const int* in_sizes, int n_in,
                              void* d_out, int out_size, void* d_ws, size_t ws_size,
                              hipStream_t stream) {
  const float* x      = (const float*)d_in[0];
  const int*   ei     = (const int*)d_in[1];
  const float* pseudo = (const float*)d_in[2];
  const int*   slices = (const int*)d_in[3];
  const float* W1     = (const float*)d_in[4];
  const float* root1  = (const float*)d_in[5];
  const float* b1     = (const float*)d_in[6];
  const float* W2     = (const float*)d_in[7];
  const float* root2  = (const float*)d_in[8];
  const float* b2     = (const float*)d_in[9];
  const float* fcw    = (const float*)d_in[10];
  const float* fcb    = (const float*)d_in[11];

  const int N = in_sizes[0] / 7;   // 51200 (multiple of 16)
  const int E = in_sizes[1] / 2;   // 819200
  const int B = in_sizes[3] - 1;   // 64
  const int* src = ei;
  const int* dst = ei + E;

  char* ws = (char*)d_ws;
  size_t off = 0;
  auto take = [&](size_t bytes) -> void* {
    void* p = ws + off;
    off += (bytes + 255) & ~(size_t)255;
    return p;
  };
  float* basis = (float*)take((size_t)E * 4 * sizeof(float));
  int*   widx  = (int*)  take((size_t)E * 4 * sizeof(int));
  float* deg   = (float*)take((size_t)N * sizeof(float));
  float* Z     = (float*)take((size_t)N * NKER * 64 * sizeof(float));  // reused both layers
  float* agg   = (float*)take((size_t)N * 64 * sizeof(float));
  float* h1    = (float*)take((size_t)N * 32 * sizeof(float));
  float* h2    = (float*)take((size_t)N * 64 * sizeof(float));
  float* g     = (float*)take((size_t)B * 64 * sizeof(float));
  (void)ws_size; (void)n_in; (void)out_size;

  const int T = 256;

  // basis + degrees (shared by both layers)
  basis_kernel<<<(E + T - 1) / T, T, 0, stream>>>(pseudo, basis, widx, E);
  hipMemsetAsync(deg, 0, (size_t)N * sizeof(float), stream);
  deg_kernel<<<(E + T - 1) / T, T, 0, stream>>>(dst, deg, E);

  // ---- layer 1: 7 -> 32 ----
  z_gemm_wmma<7, 32><<<(N / 16) * NKER, 32, 0, stream>>>(x, W1, Z);
  hipMemsetAsync(agg, 0, (size_t)N * 32 * sizeof(float), stream);
  edge_msg<32><<<(E * 4 + T - 1) / T, T, 0, stream>>>(Z, basis, widx, src, dst, agg, E);
  finalize_elu<7, 32><<<(N * 32 + T - 1) / T, T, 0, stream>>>(agg, deg, x, root1, b1, h1, N);

  // ---- layer 2: 32 -> 64 ----
  z_gemm_wmma<32, 64><<<(N / 16) * NKER, 32, 0, stream>>>(h1, W2, Z);
  hipMemsetAsync(agg, 0, (size_t)N * 64 * sizeof(float), stream);
  edge_msg<64><<<(E * 8 + T - 1) / T, T, 0, stream>>>(Z, basis, widx, src, dst, agg, E);
  finalize_elu<32, 64><<<(N * 64 + T - 1) / T, T, 0, stream>>>(agg, deg, h1, root2, b2, h2, N);

  // ---- pool + FC + log_softmax ----
  pool_kernel<<<B, 256, 0, stream>>>(h2, slices, g);
  fc_logsoftmax<<<B, 32, 0, stream>>>(g, fcw, fcb, (float*)d_out);
}